// LSA_Model_71511205478734
// MI455X (gfx1250) — compile-verified
//
#include <hip/hip_runtime.h>
#include <hip/hip_bf16.h>

#define LNUM   4
#define BATCH  4
#define SEQ    4096
#define DMODEL 256
#define KDIM   64
#define VDIM   64
#define CHUNK  256                      // S-rows per ktv block
#define NCHUNK (SEQ / CHUNK)            // 16
#define TPAD   66                       // padded k-stride of transposed LDS panel

typedef float v2f __attribute__((ext_vector_type(2)));
typedef float v8f __attribute__((ext_vector_type(8)));

__device__ __forceinline__ v8f wmma4(v2f a, v2f b, v8f c) {
  // D = A(16x4, f32) * B(4x16, f32) + C(16x16, f32)
  return __builtin_amdgcn_wmma_f32_16x16x4_f32(
      /*neg_a=*/false, a, /*neg_b=*/false, b,
      /*c_mod=*/(short)0, c, /*reuse_a=*/false, /*reuse_b=*/false);
}

// ---------------------------------------------------------------------------
// K1: Q/K/V = x[B*S, 256] @ W[256, 64].  grid=(B*S/64, 3), block=128 (4 waves,
// each wave one 16-row tile x 64 cols). Weight k-panels staged TRANSPOSED in
// LDS (sWT[col][k], padded stride) so every B fragment is one aligned
// ds_load_b64 straight into an even VGPR pair — no repack moves.
// ---------------------------------------------------------------------------
__global__ __launch_bounds__(128) void qkv_kernel(
    const float* __restrict__ x,
    const float* __restrict__ Wq, const float* __restrict__ Wk,
    const float* __restrict__ Wv,
    float* __restrict__ Qb, float* __restrict__ Kb, float* __restrict__ Vb) {
  __shared__ float sWT[64 * TPAD];      // [col][k], 16.5 KB

  const int tid   = threadIdx.x;
  const int wave  = tid >> 5;
  const int lane  = tid & 31;
  const int m     = lane & 15;
  const int kb    = (lane >> 4) << 1;    // 0 or 2
  const int rowHi = (lane >> 4) * 8;

  const int qkv = blockIdx.y;
  const float* W   = (qkv == 0) ? Wq : (qkv == 1) ? Wk : Wv;
  float*       Out = (qkv == 0) ? Qb : (qkv == 1) ? Kb : Vb;
  const int rowBase = blockIdx.x * 64 + wave * 16;

  v8f acc[4] = {};

  for (int kp = 0; kp < DMODEL; kp += 64) {
    __syncthreads();
    // cooperative transposed panel load: W rows kp..kp+63 (64x64 floats)
    const float4* src = (const float4*)(W + kp * KDIM);
    for (int i = tid; i < 1024; i += 128) {
      float4 w = src[i];
      int r  = i >> 4;                  // k within panel
      int c4 = (i & 15) << 2;           // col
      sWT[(c4 + 0) * TPAD + r] = w.x;
      sWT[(c4 + 1) * TPAD + r] = w.y;
      sWT[(c4 + 2) * TPAD + r] = w.z;
      sWT[(c4 + 3) * TPAD + r] = w.w;
    }
    __syncthreads();

    for (int k = 0; k < 64; k += 4) {
      v2f a = *(const v2f*)(x + (size_t)(rowBase + m) * DMODEL + kp + k + kb);
#pragma unroll
      for (int t = 0; t < 4; ++t) {
        v2f bf = *(const v2f*)&sWT[(t * 16 + m) * TPAD + k + kb];
        acc[t] = wmma4(a, bf, acc[t]);
      }
    }
  }

#pragma unroll
  for (int t = 0; t < 4; ++t)
#pragma unroll
    for (int j = 0; j < 8; ++j)
      Out[(size_t)(rowBase + rowHi + j) * KDIM + t * 16 + m] = acc[t][j];
}

// ---------------------------------------------------------------------------
// K2: partial M = K^T V over a 256-row S-chunk.  grid=B*NCHUNK, block=128.
// Wave w owns A-row-tile (kd in [16w,16w+16)); 4 v-tiles per wave.
// Writes Mpart[(b*NCHUNK + c)][64*64]  (no atomics -> deterministic).
// ---------------------------------------------------------------------------
__global__ __launch_bounds__(128) void ktv_kernel(
    const float* __restrict__ Kb, const float* __restrict__ Vb,
    float* __restrict__ Mpart) {
  const int tid   = threadIdx.x;
  const int wave  = tid >> 5;
  const int lane  = tid & 31;
  const int m     = lane & 15;
  const int kb    = (lane >> 4) << 1;
  const int rowHi = (lane >> 4) * 8;

  const int b = blockIdx.x / NCHUNK;
  const int c = blockIdx.x % NCHUNK;
  const float* Kp = Kb + (size_t)b * SEQ * KDIM + (size_t)c * CHUNK * KDIM;
  const float* Vp = Vb + (size_t)b * SEQ * VDIM + (size_t)c * CHUNK * VDIM;

  v8f acc[4] = {};
  for (int t = 0; t < CHUNK; t += 4) {
    v2f a;                               // A = K^T : A[kd][t] = K[t][kd]
    a.x = Kp[(t + kb)     * KDIM + wave * 16 + m];
    a.y = Kp[(t + kb + 1) * KDIM + wave * 16 + m];
#pragma unroll
    for (int vt = 0; vt < 4; ++vt) {
      v2f bf;
      bf.x = Vp[(t + kb)     * VDIM + vt * 16 + m];
      bf.y = Vp[(t + kb + 1) * VDIM + vt * 16 + m];
      acc[vt] = wmma4(a, bf, acc[vt]);
    }
  }

  float* Mo = Mpart + (size_t)blockIdx.x * (KDIM * VDIM);
#pragma unroll
  for (int vt = 0; vt < 4; ++vt)
#pragma unroll
    for (int j = 0; j < 8; ++j)
      Mo[(wave * 16 + rowHi + j) * VDIM + vt * 16 + m] = acc[vt][j];
}

// K2b: deterministic reduction of NCHUNK partials -> M[b][64*64]
__global__ void reduce_m(const float* __restrict__ Mpart, float* __restrict__ M) {
  int i = blockIdx.x * blockDim.x + threadIdx.x;   // over B*4096
  int b = i >> 12;
  int e = i & 4095;
  float s = 0.f;
  for (int c = 0; c < NCHUNK; ++c)
    s += Mpart[((size_t)(b * NCHUNK + c)) * (KDIM * VDIM) + e];
  M[i] = s;
}

// ---------------------------------------------------------------------------
// K3: Mp[b] = M[b] (64x64) @ Wp (64x256), plus a transposed copy
// MpT[b][n][k] so K4's B fragments are contiguous v2f loads.
// grid=(4 ngroups, B), block=128 (wave = m-tile).
// ---------------------------------------------------------------------------
__global__ __launch_bounds__(128) void mproj_kernel(
    const float* __restrict__ M, const float* __restrict__ Wp,
    float* __restrict__ Mp, float* __restrict__ MpT) {
  const int tid   = threadIdx.x;
  const int wave  = tid >> 5;
  const int lane  = tid & 31;
  const int m     = lane & 15;
  const int kb    = (lane >> 4) << 1;
  const int rowHi = (lane >> 4) * 8;

  const float* Mb   = M   + (size_t)blockIdx.y * KDIM * VDIM;
  float*       Mpb  = Mp  + (size_t)blockIdx.y * KDIM * DMODEL;
  float*       MpTb = MpT + (size_t)blockIdx.y * DMODEL * KDIM;
  const int nbase = blockIdx.x * 64;

  v8f acc[4] = {};
  for (int k = 0; k < VDIM; k += 4) {
    v2f a = *(const v2f*)(Mb + (wave * 16 + m) * VDIM + k + kb);
#pragma unroll
    for (int vt = 0; vt < 4; ++vt) {
      v2f bf;
      bf.x = Wp[(k + kb)     * DMODEL + nbase + vt * 16 + m];
      bf.y = Wp[(k + kb + 1) * DMODEL + nbase + vt * 16 + m];
      acc[vt] = wmma4(a, bf, acc[vt]);
    }
  }
#pragma unroll
  for (int vt = 0; vt < 4; ++vt)
#pragma unroll
    for (int j = 0; j < 8; ++j) {
      const int row = wave * 16 + rowHi + j;       // k-index of Mp
      const int col = nbase + vt * 16 + m;         // n-index of Mp
      Mpb [(size_t)row * DMODEL + col] = acc[vt][j];
      MpTb[(size_t)col * KDIM   + row] = acc[vt][j];
    }
}

// ---------------------------------------------------------------------------
// K4: out = x + Q[B*S,64] @ Mp[b][64,256].  Residual fused as the WMMA C
// initializer; B fragments from the transposed MpT (contiguous v2f).
// grid=B*S/16, block=128 (wave = 64-col group).
// ---------------------------------------------------------------------------
__global__ __launch_bounds__(128) void out_kernel(
    const float* __restrict__ x, const float* __restrict__ Qb,
    const float* __restrict__ MpT, float* __restrict__ out) {
  const int tid   = threadIdx.x;
  const int wave  = tid >> 5;
  const int lane  = tid & 31;
  const int m     = lane & 15;
  const int kb    = (lane >> 4) << 1;
  const int rowHi = (lane >> 4) * 8;

  const int rbase = blockIdx.x * 16;
  const int b     = rbase / SEQ;
  const int nbase = wave * 64;
  const float* MpTb = MpT + (size_t)b * DMODEL * KDIM;

  v8f acc[4];
#pragma unroll
  for (int vt = 0; vt < 4; ++vt)
#pragma unroll
    for (int j = 0; j < 8; ++j)
      acc[vt][j] = x[(size_t)(rbase + rowHi + j) * DMODEL + nbase + vt * 16 + m];

  for (int k = 0; k < KDIM; k += 4) {
    v2f a = *(const v2f*)(Qb + (size_t)(rbase + m) * KDIM + k + kb);
#pragma unroll
    for (int vt = 0; vt < 4; ++vt) {
      v2f bf = *(const v2f*)(MpTb + (size_t)(nbase + vt * 16 + m) * KDIM + k + kb);
      acc[vt] = wmma4(a, bf, acc[vt]);
    }
  }

#pragma unroll
  for (int vt = 0; vt < 4; ++vt)
#pragma unroll
    for (int j = 0; j < 8; ++j)
      out[(size_t)(rbase + rowHi + j) * DMODEL + nbase + vt * 16 + m] = acc[vt][j];
}

// ---------------------------------------------------------------------------
extern "C" void kernel_launch(void* const* d_in, const int* in_sizes, int n_in,
                              void* d_out, int out_size, void* d_ws, size_t ws_size,
                              hipStream_t stream) {
  (void)in_sizes; (void)n_in; (void)out_size; (void)ws_size;
  const float* x  = (const float*)d_in[0];
  const float* Wq = (const float*)d_in[1];
  const float* Wk = (const float*)d_in[2];
  const float* Wv = (const float*)d_in[3];
  const float* Wp = (const float*)d_in[4];
  float* out = (float*)d_out;

  float* Qb    = (float*)d_ws;                                 // B*S*64
  float* Kb    = Qb    + (size_t)BATCH * SEQ * KDIM;           // B*S*64
  float* Vb    = Kb    + (size_t)BATCH * SEQ * KDIM;           // B*S*64
  float* Mpart = Vb    + (size_t)BATCH * SEQ * VDIM;           // B*NCHUNK*64*64
  float* M     = Mpart + (size_t)BATCH * NCHUNK * KDIM * VDIM; // B*64*64
  float* Mp    = M     + (size_t)BATCH * KDIM * VDIM;          // B*64*256
  float* MpT   = Mp    + (size_t)BATCH * KDIM * DMODEL;        // B*256*64
  float* xbuf  = MpT   + (size_t)BATCH * DMODEL * KDIM;        // B*S*D

  for (int l = 0; l < LNUM; ++l) {
    const float* xin  = (l == 0)        ? x   : xbuf;
    float*       xout = (l == LNUM - 1) ? out : xbuf;
    const float* wq = Wq + (size_t)l * DMODEL * KDIM;
    const float* wk = Wk + (size_t)l * DMODEL * KDIM;
    const float* wv = Wv + (size_t)l * DMODEL * VDIM;
    const float* wp = Wp + (size_t)l * VDIM * DMODEL;

    qkv_kernel<<<dim3(BATCH * SEQ / 64, 3), 128, 0, stream>>>(
        xin, wq, wk, wv, Qb, Kb, Vb);
    ktv_kernel<<<BATCH * NCHUNK, 128, 0, stream>>>(Kb, Vb, Mpart);
    reduce_m<<<(BATCH * KDIM * VDIM) / 256, 256, 0, stream>>>(Mpart, M);
    mproj_kernel<<<dim3(4, BATCH), 128, 0, stream>>>(M, wp, Mp, MpT);
    out_kernel<<<BATCH * SEQ / 16, 128, 0, stream>>>(xin, Qb, MpT, xout);
  }
}